// MyNet_45148696215616
// MI455X (gfx1250) — compile-verified
//
#include <hip/hip_runtime.h>
#include <math.h>

typedef float v2f __attribute__((ext_vector_type(2)));
typedef float v8f __attribute__((ext_vector_type(8)));

__device__ __forceinline__ float lrelu(float x) { return x >= 0.f ? x : 0.2f * x; }

// sign-aware float atomic max via integer punning (works for mixed-sign values)
__device__ __forceinline__ void atomicMaxF(float* addr, float v) {
    if (v >= 0.f) atomicMax((int*)addr, __float_as_int(v));
    else          atomicMin((unsigned int*)addr, __float_as_uint(v));
}

// ---------------------------------------------------------------------------
// GEMM1: h[N,64] = x[N,128] @ W[128,64] + b   (f32 WMMA 16x16x4)
// block = 128 threads = 4 waves; each wave owns one 16-col tile of a 16-row tile
// ---------------------------------------------------------------------------
__global__ void gemm1_wmma(const float* __restrict__ x, const float* __restrict__ W,
                           const float* __restrict__ b, float* __restrict__ h, int N) {
    const int K = 128, F = 64;
    int rowTile = blockIdx.x * 16;
    int wave = threadIdx.x >> 5;
    int lane = threadIdx.x & 31;
    int half = lane >> 4;      // 0: K pair (k,k+1) / M rows r; 1: (k+2,k+3) / r+8
    int l    = lane & 15;
    int n0   = wave * 16;
    int arow = rowTile + l; if (arow >= N) arow = N - 1;

    v8f acc = {};
    for (int k = 0; k < K; k += 4) {
        // A frag: lane group holds K = k+2*half, k+2*half+1 for row M=l
        const float* ap = x + (size_t)arow * K + k + half * 2;
        v2f a; a.x = ap[0]; a.y = ap[1];
        // B frag: VGPR0 = K rows {k, k+2}, VGPR1 = {k+1, k+3}, N = n0+l per lane
        const float* bp = W + (size_t)(k + half * 2) * F + n0 + l;
        v2f bb; bb.x = bp[0]; bb.y = bp[F];
        acc = __builtin_amdgcn_wmma_f32_16x16x4_f32(false, a, false, bb,
                                                    (short)0, acc, false, false);
    }
    float bias = b[n0 + l];
    for (int r = 0; r < 8; ++r) {
        int m = rowTile + r + half * 8;
        if (m < N) h[(size_t)m * F + n0 + l] = acc[r] + bias;
    }
}

// ---------------------------------------------------------------------------
// GEMM2: h2[N,16] = (hagg[N,64]/cnt) @ W[64,16] + b   (mean fused into A load)
// block = 32 threads = 1 wave, single 16-col tile
// ---------------------------------------------------------------------------
__global__ void gemm2_wmma(const float* __restrict__ hin, const float* __restrict__ cnt,
                           const float* __restrict__ W, const float* __restrict__ b,
                           float* __restrict__ h2, int N) {
    const int K = 64, F = 16;
    int rowTile = blockIdx.x * 16;
    int lane = threadIdx.x & 31;
    int half = lane >> 4;
    int l    = lane & 15;
    int arow = rowTile + l; if (arow >= N) arow = N - 1;
    float rc = 1.0f / cnt[arow];

    v8f acc = {};
    for (int k = 0; k < K; k += 4) {
        const float* ap = hin + (size_t)arow * K + k + half * 2;
        v2f a; a.x = ap[0] * rc; a.y = ap[1] * rc;
        const float* bp = W + (size_t)(k + half * 2) * F + l;
        v2f bb; bb.x = bp[0]; bb.y = bp[F];
        acc = __builtin_amdgcn_wmma_f32_16x16x4_f32(false, a, false, bb,
                                                    (short)0, acc, false, false);
    }
    float bias = b[l];
    for (int r = 0; r < 8; ++r) {
        int m = rowTile + r + half * 8;
        if (m < N) h2[(size_t)m * F + l] = acc[r] + bias;
    }
}

// ---------------------------------------------------------------------------
// Per-node attention scores + segment-max init with the self-loop value
// ---------------------------------------------------------------------------
__global__ void node_scores(const float* __restrict__ h, int F, const float* __restrict__ a,
                            float* __restrict__ ssrc, float* __restrict__ sdst,
                            float* __restrict__ m, int N) {
    int i = blockIdx.x * blockDim.x + threadIdx.x;
    if (i >= N) return;
    const float* hp = h + (size_t)i * F;
    float s0 = 0.f, s1 = 0.f;
    for (int f = 0; f < F; ++f) { float v = hp[f]; s0 += v * a[f]; s1 += v * a[F + f]; }
    ssrc[i] = s0; sdst[i] = s1;
    m[i] = lrelu(s0 + s1);   // every node has a self-loop edge (i -> i)
}

__global__ void edge_max(const int* __restrict__ row, const int* __restrict__ col,
                         const float* __restrict__ ssrc, const float* __restrict__ sdst,
                         float* __restrict__ m, int E) {
    int e = blockIdx.x * blockDim.x + threadIdx.x;
    if (e >= E) return;
    int r = row[e], c = col[e];
    atomicMaxF(&m[c], lrelu(ssrc[r] + sdst[c]));
}

__global__ void node_den_init(const float* __restrict__ ssrc, const float* __restrict__ sdst,
                              const float* __restrict__ m, float* __restrict__ den,
                              float* __restrict__ cnt, int N) {
    int i = blockIdx.x * blockDim.x + threadIdx.x;
    if (i >= N) return;
    den[i] = expf(lrelu(ssrc[i] + sdst[i]) - m[i]);   // self-loop term
    if (cnt) cnt[i] = 1.0f;
}

__global__ void edge_denom(const int* __restrict__ row, const int* __restrict__ col,
                           const float* __restrict__ ssrc, const float* __restrict__ sdst,
                           const float* __restrict__ m, float* __restrict__ den,
                           float* __restrict__ cnt, int E) {
    int e = blockIdx.x * blockDim.x + threadIdx.x;
    if (e >= E) return;
    int r = row[e], c = col[e];
    float ex = expf(lrelu(ssrc[r] + sdst[c]) - m[c]);
    atomicAdd(&den[c], ex);
    if (cnt) atomicAdd(&cnt[c], 1.0f);
}

__global__ void edge_norm(const int* __restrict__ row, const int* __restrict__ col,
                          const float* __restrict__ ssrc, const float* __restrict__ sdst,
                          const float* __restrict__ m, const float* __restrict__ den,
                          float* __restrict__ norm, int E) {
    int e = blockIdx.x * blockDim.x + threadIdx.x;
    if (e >= E) return;
    int r = row[e], c = col[e];
    norm[e] = expf(lrelu(ssrc[r] + sdst[c]) - m[c]) / den[c];
}

// hagg[i,f] initialized with the self-loop message (also defines every element)
template <int F>
__global__ void agg_init(const float* __restrict__ h, const float* __restrict__ ssrc,
                         const float* __restrict__ sdst, const float* __restrict__ m,
                         const float* __restrict__ den, float* __restrict__ hagg, int N) {
    int t = blockIdx.x * blockDim.x + threadIdx.x;
    if (t >= N * F) return;
    int i = t >> (F == 64 ? 6 : 4);
    float w = expf(lrelu(ssrc[i] + sdst[i]) - m[i]) / den[i];
    hagg[t] = w * h[t];
}

template <int F>
__global__ void edge_agg(const int* __restrict__ row, const int* __restrict__ col,
                         const float* __restrict__ norm, const float* __restrict__ h,
                         float* __restrict__ hagg, int E) {
    long long t = (long long)blockIdx.x * blockDim.x + threadIdx.x;
    if (t >= (long long)E * F) return;
    int e = (int)(t >> (F == 64 ? 6 : 4));
    int f = (int)(t & (F - 1));
    int r = row[e], c = col[e];
    atomicAdd(&hagg[(size_t)c * F + f], norm[e] * h[(size_t)r * F + f]);
}

__global__ void final_logsoftmax(const float* __restrict__ hagg2, const float* __restrict__ cnt,
                                 float* __restrict__ out, int N) {
    int i = blockIdx.x * blockDim.x + threadIdx.x;
    if (i >= N) return;
    float rc = 1.0f / cnt[i];
    float v[16];
    float mx = -INFINITY;
    for (int c = 0; c < 16; ++c) { v[c] = hagg2[(size_t)i * 16 + c] * rc; mx = fmaxf(mx, v[c]); }
    float s = 0.f;
    for (int c = 0; c < 16; ++c) s += expf(v[c] - mx);
    float lse = mx + logf(s);
    for (int c = 0; c < 16; ++c) out[(size_t)i * 16 + c] = v[c] - lse;
}

// ---------------------------------------------------------------------------
extern "C" void kernel_launch(void* const* d_in, const int* in_sizes, int n_in,
                              void* d_out, int out_size, void* d_ws, size_t ws_size,
                              hipStream_t stream) {
    (void)n_in; (void)out_size; (void)ws_size;
    const float* x  = (const float*)d_in[0];
    const int*   ei = (const int*)d_in[1];
    const float* W1 = (const float*)d_in[2];
    const float* b1 = (const float*)d_in[3];
    const float* a1 = (const float*)d_in[4];
    const float* W2 = (const float*)d_in[5];
    const float* b2 = (const float*)d_in[6];
    const float* a2 = (const float*)d_in[7];
    float* out = (float*)d_out;

    int N = in_sizes[0] / 128;
    int E = in_sizes[1] / 2;
    const int* row = ei;         // source nodes
    const int* col = ei + E;     // target nodes

    float* ws = (float*)d_ws;
    size_t nf = (size_t)N;
    // phase 1 layout
    float* h1    = ws;               // N*64
    float* hagg1 = ws + nf * 64;     // N*64
    float* s1s   = ws + nf * 128;    // N
    float* s1d   = s1s + nf;         // N
    float* m1    = s1s + 2 * nf;     // N
    float* den1  = s1s + 3 * nf;     // N
    float* cnt   = s1s + 4 * nf;     // N (shared: same graph both layers)
    float* norm1 = s1s + 5 * nf;     // E
    // phase 2 layout (reuses dead h1 + norm1 regions; hagg1/cnt stay live)
    float* h2    = ws;               // N*16
    float* hagg2 = ws + nf * 16;     // N*16
    float* s2s   = ws + nf * 32;     // N
    float* s2d   = ws + nf * 33;     // N
    float* m2    = ws + nf * 34;     // N
    float* den2  = ws + nf * 35;     // N
    float* norm2 = norm1;            // E

    int nodeBlocks = (N + 255) / 256;
    int edgeBlocks = (E + 255) / 256;
    int rowTiles   = (N + 15) / 16;

    // ---- layer 1 (F = 64) ----
    gemm1_wmma<<<rowTiles, 128, 0, stream>>>(x, W1, b1, h1, N);
    node_scores<<<nodeBlocks, 256, 0, stream>>>(h1, 64, a1, s1s, s1d, m1, N);
    edge_max<<<edgeBlocks, 256, 0, stream>>>(row, col, s1s, s1d, m1, E);
    node_den_init<<<nodeBlocks, 256, 0, stream>>>(s1s, s1d, m1, den1, cnt, N);
    edge_denom<<<edgeBlocks, 256, 0, stream>>>(row, col, s1s, s1d, m1, den1, cnt, E);
    edge_norm<<<edgeBlocks, 256, 0, stream>>>(row, col, s1s, s1d, m1, den1, norm1, E);
    agg_init<64><<<(int)((nf * 64 + 255) / 256), 256, 0, stream>>>(h1, s1s, s1d, m1, den1, hagg1, N);
    edge_agg<64><<<(int)(((size_t)E * 64 + 255) / 256), 256, 0, stream>>>(row, col, norm1, h1, hagg1, E);

    // ---- layer 2 (F = 16); mean of layer 1 fused into GEMM2 A-fragment ----
    gemm2_wmma<<<rowTiles, 32, 0, stream>>>(hagg1, cnt, W2, b2, h2, N);
    node_scores<<<nodeBlocks, 256, 0, stream>>>(h2, 16, a2, s2s, s2d, m2, N);
    edge_max<<<edgeBlocks, 256, 0, stream>>>(row, col, s2s, s2d, m2, E);
    node_den_init<<<nodeBlocks, 256, 0, stream>>>(s2s, s2d, m2, den2, nullptr, N);
    edge_denom<<<edgeBlocks, 256, 0, stream>>>(row, col, s2s, s2d, m2, den2, nullptr, E);
    edge_norm<<<edgeBlocks, 256, 0, stream>>>(row, col, s2s, s2d, m2, den2, norm2, E);
    agg_init<16><<<(int)((nf * 16 + 255) / 256), 256, 0, stream>>>(h2, s2s, s2d, m2, den2, hagg2, N);
    edge_agg<16><<<(int)(((size_t)E * 16 + 255) / 256), 256, 0, stream>>>(row, col, norm2, h2, hagg2, E);

    // ---- mean + log_softmax ----
    final_logsoftmax<<<nodeBlocks, 256, 0, stream>>>(hagg2, cnt, out, N);
}